// TripletAttention_33663953666883
// MI455X (gfx1250) — compile-verified
//
#include <hip/hip_runtime.h>
#include <math.h>

// ---------------------------------------------------------------------------
// TripletAttention for MI455X (gfx1250).
//  * All GEMMs through V_WMMA_F32_16X16X4_F32 (fp32, matches reference).
//  * Weights pre-transposed (+head-major column permutation) so every WMMA
//    needs exactly one global b64 (B frag) + one LDS b64 (A frag).
//  * LDS staging via Tensor Data Mover (tensor_load_to_lds) with HW padding
//    for conflict-free strides; s_wait_tensorcnt + barrier for ordering.
// ---------------------------------------------------------------------------

#define NSEQ   160
#define CDIM   256
#define HDIM   8
#define M_TOT  (NSEQ * NSEQ)          // 25600 flattened (i,j) rows
#define QK_SCALE 0.17677669529663687f // 1/sqrt(32), folded into Wq/bq in prep
#define LN_EPS   1e-5f

typedef float v2f __attribute__((ext_vector_type(2)));
typedef float v8f __attribute__((ext_vector_type(8)));
typedef unsigned int u32x4 __attribute__((ext_vector_type(4)));
typedef int i32x4 __attribute__((ext_vector_type(4)));
typedef int i32x8 __attribute__((ext_vector_type(8)));

#if defined(__has_builtin)
#if __has_builtin(__builtin_amdgcn_tensor_load_to_lds)
#define USE_TDM 1
#endif
#endif
#ifndef USE_TDM
#define USE_TDM 0
#endif
// Toolchain arity heuristic: therock headers ship the 6-arg builtin.
#if __has_include(<hip/amd_detail/amd_gfx1250_TDM.h>)
#define TDM_SIX_ARGS 1
#else
#define TDM_SIX_ARGS 0
#endif

// D(16x16 f32) = A(16x4 f32) * B(4x16 f32) + C   (CDNA5 ISA 7.12.2 layouts)
// A frag: m = lane&15, k = 2*(lane>>4)+{0,1};  B frag: n = lane&15, same k.
// C/D frag: VGPR v -> m = v + 8*(lane>>4), n = lane&15.
__device__ __forceinline__ v8f wmma_f32(v2f a, v2f b, v8f c) {
  return __builtin_amdgcn_wmma_f32_16x16x4_f32(
      false, a, false, b, (short)0, c, false, false);
}

#if USE_TDM
__device__ __forceinline__ unsigned lds_off(const void* p) {
  // LDS aperture flat address truncates to the LDS byte address (ISA 10.2).
  return (unsigned)(unsigned long long)p;
}
// 2D tile load Global->LDS via TDM. data_size=4B. Optional LDS padding of
// 1 dword every (1<<(pad_code+1)) dwords (pad_code 4 -> every 32, 7 -> 256).
__device__ __forceinline__ void tdm_load_2d(unsigned lds_addr, const void* gptr,
                                            unsigned d0, unsigned d1,
                                            unsigned long long row_stride,
                                            unsigned pad_code, bool pad_en) {
  unsigned long long ga = (unsigned long long)gptr;
  u32x4 g0;
  g0[0] = 1u;                                        // count=1
  g0[1] = lds_addr;                                  // lds_addr [63:32]
  g0[2] = (unsigned)(ga & 0xffffffffu);              // global_addr lo
  g0[3] = (unsigned)((ga >> 32) & 0x1ffffffu) | (2u << 30); // addr hi | type=2
  unsigned w0 = (2u << 16);                          // data_size = 4B
  if (pad_en) w0 |= (1u << 20) | (pad_code << 22);   // pad_amount=0 => 1 dword
  i32x8 g1;
  g1[0] = (int)w0;
  g1[1] = (int)((d0 & 0xffffu) << 16);               // tensor_dim0[15:0]
  g1[2] = (int)(((d0 >> 16) & 0xffffu) | ((d1 & 0xffffu) << 16)); // d0 hi|d1 lo
  g1[3] = (int)(((d1 >> 16) & 0xffffu) | ((d0 & 0xffffu) << 16)); // d1 hi|tile_dim0
  g1[4] = (int)(d1 & 0xffffu);                       // tile_dim1 (tile_dim2=0)
  g1[5] = (int)(unsigned)(row_stride & 0xffffffffu); // tensor_dim0_stride lo32
  g1[6] = (int)(unsigned)((row_stride >> 32) & 0xffffu); // stride hi16
  g1[7] = 0;
  i32x4 z4 = {0, 0, 0, 0};
#if TDM_SIX_ARGS
  i32x8 z8 = {0, 0, 0, 0, 0, 0, 0, 0};
  __builtin_amdgcn_tensor_load_to_lds(g0, g1, z4, z4, z8, 0);
#else
  __builtin_amdgcn_tensor_load_to_lds(g0, g1, z4, z4, 0);
#endif
}
#endif  // USE_TDM

// ---------------------------------------------------------------------------
// Weight prep kernels (run once per launch; tiny).
// qkv weights: transpose to Wt[c'][256] with head-major column permutation
//   c' = (s*8+h)*32 + d  <->  logical c = s*256 + d*8 + h, fold QK_SCALE
//   into the q (s==0) columns and bias.
// ---------------------------------------------------------------------------
__global__ __launch_bounds__(256) void prep_qkv_w(const float* __restrict__ W,
                                                  const float* __restrict__ b,
                                                  float* __restrict__ Wt,
                                                  float* __restrict__ bp) {
  int tid = blockIdx.x * 256 + threadIdx.x;  // tid = c'*256 + k
  int cp = tid >> 8, k = tid & 255;
  int p = cp >> 5, d = cp & 31;
  int s = p >> 3, h = p & 7;
  int c = s * 256 + d * 8 + h;
  float sc = (s == 0) ? QK_SCALE : 1.0f;
  Wt[tid] = W[(size_t)k * 768 + c] * sc;
  if (k == 0) bp[cp] = b[c] * sc;
}
__global__ __launch_bounds__(256) void prep_eg_w(const float* __restrict__ W,
                                                 float* __restrict__ Wt) {
  int tid = blockIdx.x * 256 + threadIdx.x;  // tid = c*256 + k, c < 16
  int c = tid >> 8, k = tid & 255;
  Wt[tid] = W[(size_t)k * 16 + c];
}
// Wo rows permuted with the SAME head-major K-order the attention kernels
// emit: row' = (branch*8+h)*32 + d  <->  logical k = branch*256 + d*8 + h.
__global__ __launch_bounds__(256) void prep_wo(const float* __restrict__ Wo,
                                               float* __restrict__ Wt) {
  int tid = blockIdx.x * 256 + threadIdx.x;  // tid = n*512 + c'
  int n = tid >> 9, cp = tid & 511;
  int p = cp >> 5, d = cp & 31;
  int br = p >> 3, h = p & 7;
  int k = br * 256 + d * 8 + h;
  Wt[tid] = Wo[(size_t)k * 256 + n];
}

// ---------------------------------------------------------------------------
// LayerNorm: one wave32 per 256-wide row.
// ---------------------------------------------------------------------------
__global__ __launch_bounds__(256) void ln_kernel(
    const float* __restrict__ e, const float* __restrict__ g,
    const float* __restrict__ b, float* __restrict__ out) {
  int row  = blockIdx.x * 8 + (threadIdx.x >> 5);
  int lane = threadIdx.x & 31;
  const float4* x4 = (const float4*)(e + (size_t)row * CDIM);
  float4 v0 = x4[lane];
  float4 v1 = x4[lane + 32];
  float s = (v0.x + v0.y) + (v0.z + v0.w) + (v1.x + v1.y) + (v1.z + v1.w);
#pragma unroll
  for (int o = 16; o; o >>= 1) s += __shfl_xor(s, o, 32);
  float mu = s * (1.0f / 256.0f);
  float a0 = v0.x - mu, a1 = v0.y - mu, a2 = v0.z - mu, a3 = v0.w - mu;
  float a4 = v1.x - mu, a5 = v1.y - mu, a6 = v1.z - mu, a7 = v1.w - mu;
  float q = a0*a0 + a1*a1 + a2*a2 + a3*a3 + a4*a4 + a5*a5 + a6*a6 + a7*a7;
#pragma unroll
  for (int o = 16; o; o >>= 1) q += __shfl_xor(q, o, 32);
  float rstd = rsqrtf(q * (1.0f / 256.0f) + LN_EPS);
  const float4* g4 = (const float4*)g;
  const float4* b4 = (const float4*)b;
  float4 ga = g4[lane], gb = g4[lane + 32];
  float4 ba = b4[lane], bb = b4[lane + 32];
  float4 o0, o1;
  o0.x = a0 * rstd * ga.x + ba.x;  o0.y = a1 * rstd * ga.y + ba.y;
  o0.z = a2 * rstd * ga.z + ba.z;  o0.w = a3 * rstd * ga.w + ba.w;
  o1.x = a4 * rstd * gb.x + bb.x;  o1.y = a5 * rstd * gb.y + bb.y;
  o1.z = a6 * rstd * gb.z + bb.z;  o1.w = a7 * rstd * gb.w + bb.w;
  float4* y4 = (float4*)(out + (size_t)row * CDIM);
  y4[lane]      = o0;
  y4[lane + 32] = o1;
}

// ---------------------------------------------------------------------------
// WMMA GEMM: out[M,N] = A[M,K] @ Wt[N,K]^T + bias.  Block = 8 waves,
// 16 rows x 128 cols; wave = one 16x16 tile, K-loop step 4.
// A tile staged in LDS by TDM with 1-dword pad every 256 dwords
//   -> element (r,c) at r*ldp + c + (c>>8),  ldp = K + K/256  (gcd(ldp,64)=1 for K=256).
// EG_T: store transposed (out[c*M_TOT + row]) for the eg buffers.
// ---------------------------------------------------------------------------
template <bool EG_T>
__global__ __launch_bounds__(256) void gemm_kernel(
    const float* __restrict__ A, const float* __restrict__ Wt,
    const float* __restrict__ bias, float* __restrict__ out, int K, int N) {
  extern __shared__ float As[];
  const int ldp  = K + (K >> 8);
  const int wave = threadIdx.x >> 5;
  const int lane = threadIdx.x & 31;
  const int half = lane >> 4;
  const int nl   = lane & 15;
  const size_t arow = (size_t)blockIdx.x * 16;
  const int    ncol = (blockIdx.y * 8 + wave) * 16;

#if USE_TDM
  if (threadIdx.x < 32) {  // wave 0 issues the tile DMA
    tdm_load_2d(lds_off(As), A + arow * (size_t)K, (unsigned)K, 16u,
                (unsigned long long)K, 7u, true);
    __builtin_amdgcn_s_wait_tensorcnt(0);
  }
#else
  for (int idx = threadIdx.x; idx < 16 * K; idx += 256) {
    int r = idx / K, c = idx - r * K;
    As[r * ldp + c + (c >> 8)] = A[(arow + r) * (size_t)K + c];
  }
#endif
  __syncthreads();

  if (ncol < N) {  // wave-uniform: EXEC all-ones inside
    v8f acc = {};
    const float* wrow = Wt + (size_t)(ncol + nl) * K;
    for (int kk = 0; kk < K; kk += 4) {
      int kb = kk + 2 * half;
      int ci = kb + (kb >> 8);
      v2f a, b;
      a.x = As[nl * ldp + ci];
      a.y = As[nl * ldp + ci + 1];
      float2 bv = *(const float2*)(wrow + kb);  // one global_load_b64
      b.x = bv.x;
      b.y = bv.y;
      acc = wmma_f32(a, b, acc);
    }
    float bb = bias[ncol + nl];
#pragma unroll
    for (int v = 0; v < 8; ++v) {
      int m = v + 8 * half;
      if (EG_T)
        out[(size_t)(ncol + nl) * M_TOT + (arow + m)] = acc[v] + bb;
      else
        out[(arow + m) * (size_t)N + ncol + nl] = acc[v] + bb;
    }
  }
}

// ---------------------------------------------------------------------------
// Attention: one (j,h) problem per workgroup, grid = 160*8.
// qkv layout (head-major, produced by the permuted projection):
//   qkv[row][p*32 + d], p = s*8+h, s in {q,k,v}  -> dense 32-float rows.
// LDS: Qs/Ks/Vs [160][33] (TDM pad every 32 dwords) + S [160][161]. ~166 KB
// of the 320 KB/WGP CDNA5 LDS -> full-S formulation, no streaming.
// in : S[i,k] = Q[i]·K[k] + E[i,k] + m[i,k], K/V rows j*160+k
// out: S[i,k] = Q[i]·K[k] + E[k,i] + m[k,i], K/V rows k*160+j
// then A = softmax_k(S) * sigmoid(G+m), O = A@V, stored head-major into Va.
// ---------------------------------------------------------------------------
template <bool OUT>
__global__ __launch_bounds__(256) void attn_kernel(
    const float* __restrict__ qkv, const float* __restrict__ egT,
    const float* __restrict__ mask, float* __restrict__ Va) {
  const int j    = blockIdx.x >> 3;
  const int h    = blockIdx.x & 7;
  const int wave = threadIdx.x >> 5;
  const int lane = threadIdx.x & 31;
  const int half = lane >> 4;
  const int nl   = lane & 15;

  extern __shared__ float smem[];
  float* Qs = smem;                  // [160][33]
  float* Ks = Qs + 160 * 33;
  float* Vs = Ks + 160 * 33;
  float* Ss = Vs + 160 * 33;         // [160][161]

  const size_t qstride = (size_t)NSEQ * 768;               // rows i*160+j
  const size_t kstride = OUT ? (size_t)NSEQ * 768 : 768;   // rows per branch
  const float* qbase = qkv + (size_t)j * 768 + (size_t)h * 32;
  const float* kbase = (OUT ? qkv + (size_t)j * 768
                            : qkv + (size_t)j * NSEQ * 768) + (size_t)(8 + h) * 32;
  const float* vbase = kbase + 256;  // v plane is +8 planes = +256 floats

#if USE_TDM
  if (threadIdx.x < 32) {
    tdm_load_2d(lds_off(Qs), qbase, 32u, 160u, (unsigned long long)qstride, 4u, true);
    tdm_load_2d(lds_off(Ks), kbase, 32u, 160u, (unsigned long long)kstride, 4u, true);
    tdm_load_2d(lds_off(Vs), vbase, 32u, 160u, (unsigned long long)kstride, 4u, true);
    __builtin_amdgcn_s_wait_tensorcnt(0);
  }
#else
  for (int idx = threadIdx.x; idx < 160 * 8; idx += 256) {
    int r = idx >> 3, c4 = (idx & 7) * 4;
    float4 qv = *(const float4*)(qbase + r * qstride + c4);
    float4 kv = *(const float4*)(kbase + r * kstride + c4);
    float4 vv = *(const float4*)(vbase + r * kstride + c4);
    *(float4*)&Qs[r * 33 + c4] = qv;  // 4B-aligned rows; compiler splits stores
    *(float4*)&Ks[r * 33 + c4] = kv;
    *(float4*)&Vs[r * 33 + c4] = vv;
  }
#endif
  __syncthreads();

  // S = Q @ K^T + E + mask. 10x10 tiles of 16x16 over 8 waves.
  for (int t = wave; t < 100; t += 8) {
    int mi = t / 10, ki = t - mi * 10;
    v8f acc = {};
#pragma unroll
    for (int kk = 0; kk < 32; kk += 4) {
      int kb = kk + 2 * half;
      v2f a, b;
      a.x = Qs[(mi * 16 + nl) * 33 + kb];
      a.y = Qs[(mi * 16 + nl) * 33 + kb + 1];
      b.x = Ks[(ki * 16 + nl) * 33 + kb];
      b.y = Ks[(ki * 16 + nl) * 33 + kb + 1];
      acc = wmma_f32(a, b, acc);
    }
#pragma unroll
    for (int v = 0; v < 8; ++v) {
      int irow = mi * 16 + v + 8 * half;
      int kcol = ki * 16 + nl;
      size_t epos = OUT ? (size_t)(kcol * NSEQ + irow)
                        : (size_t)(irow * NSEQ + kcol);
      float Ev = egT[(size_t)h * M_TOT + epos];
      float Mv = mask[epos * 8 + h];
      Ss[irow * 161 + kcol] = acc[v] + Ev + Mv;
    }
  }
  __syncthreads();

  // Row softmax over k, times sigmoid gate. One wave per row.
  for (int r = wave; r < 160; r += 8) {
    float vals[5];
    float mx = -INFINITY;
#pragma unroll
    for (int c = 0; c < 5; ++c) {
      vals[c] = Ss[r * 161 + c * 32 + lane];
      mx = fmaxf(mx, vals[c]);
    }
#pragma unroll
    for (int o = 16; o; o >>= 1) mx = fmaxf(mx, __shfl_xor(mx, o, 32));
    float sum = 0.0f;
#pragma unroll
    for (int c = 0; c < 5; ++c) { vals[c] = expf(vals[c] - mx); sum += vals[c]; }
#pragma unroll
    for (int o = 16; o; o >>= 1) sum += __shfl_xor(sum, o, 32);
    float inv = 1.0f / sum;
#pragma unroll
    for (int c = 0; c < 5; ++c) {
      int kcol = c * 32 + lane;
      size_t epos = OUT ? (size_t)(kcol * NSEQ + r) : (size_t)(r * NSEQ + kcol);
      float Gv = egT[(size_t)(8 + h) * M_TOT + epos];
      float Mv = mask[epos * 8 + h];
      float gate = 1.0f / (1.0f + expf(-(Gv + Mv)));
      Ss[r * 161 + kcol] = vals[c] * inv * gate;
    }
  }
  __syncthreads();

  // O = A @ V. 10 (m) x 2 (d) tiles, K = 160. Head-major coalesced stores.
  for (int t = wave; t < 20; t += 8) {
    int mi = t >> 1, di = t & 1;
    v8f acc = {};
    for (int kk = 0; kk < 160; kk += 4) {
      int kb = kk + 2 * half;
      v2f a, b;
      a.x = Ss[(mi * 16 + nl) * 161 + kb];
      a.y = Ss[(mi * 16 + nl) * 161 + kb + 1];
      b.x = Vs[kb * 33 + di * 16 + nl];
      b.y = Vs[(kb + 1) * 33 + di * 16 + nl];
      acc = wmma_f32(a, b, acc);
    }
#pragma unroll
    for (int v = 0; v < 8; ++v) {
      int irow = mi * 16 + v + 8 * half;
      int d    = di * 16 + nl;
      size_t orow = (size_t)(irow * NSEQ + j);
      Va[orow * 512 + (size_t)((OUT ? 8 : 0) + h) * 32 + d] = acc[v];
    }
  }
}

// ---------------------------------------------------------------------------
// Host launcher.
// ---------------------------------------------------------------------------
extern "C" void kernel_launch(void* const* d_in, const int* in_sizes, int n_in,
                              void* d_out, int out_size, void* d_ws, size_t ws_size,
                              hipStream_t stream) {
  (void)in_sizes; (void)n_in; (void)out_size; (void)ws_size;
  const float* e        = (const float*)d_in[0];
  const float* mask     = (const float*)d_in[1];
  const float* ln_g     = (const float*)d_in[2];
  const float* ln_b     = (const float*)d_in[3];
  const float* Wqkv_in  = (const float*)d_in[4];
  const float* bqkv_in  = (const float*)d_in[5];
  const float* Weg_in   = (const float*)d_in[6];
  const float* beg_in   = (const float*)d_in[7];
  const float* Wqkv_out = (const float*)d_in[8];
  const float* bqkv_out = (const float*)d_in[9];
  const float* Weg_out  = (const float*)d_in[10];
  const float* beg_out  = (const float*)d_in[11];
  const float* Wo       = (const float*)d_in[12];
  const float* bo       = (const float*)d_in[13];
  float* out = (float*)d_out;

  float* ws       = (float*)d_ws;
  float* e_ln     = ws;  ws += (size_t)M_TOT * CDIM;
  float* qkv_in   = ws;  ws += (size_t)M_TOT * 768;
  float* qkv_out  = ws;  ws += (size_t)M_TOT * 768;
  float* eg_in_T  = ws;  ws += (size_t)M_TOT * 16;   // [16][M]
  float* eg_out_T = ws;  ws += (size_t)M_TOT * 16;
  float* VaBuf    = ws;  ws += (size_t)M_TOT * 512;  // head-major cols
  float* WtQ_in   = ws;  ws += (size_t)768 * 256;
  float* WtQ_out  = ws;  ws += (size_t)768 * 256;
  float* WtE_in   = ws;  ws += (size_t)16 * 256;
  float* WtE_out  = ws;  ws += (size_t)16 * 256;
  float* WtO      = ws;  ws += (size_t)256 * 512;
  float* bpQ_in   = ws;  ws += 768;
  float* bpQ_out  = ws;  // += 768

  const size_t lds256 = (size_t)16 * 257 * sizeof(float);
  const size_t lds512 = (size_t)16 * 514 * sizeof(float);
  const size_t ldsAtt = (size_t)(3 * 160 * 33 + 160 * 161) * sizeof(float);

  hipFuncSetAttribute((const void*)attn_kernel<false>,
                      hipFuncAttributeMaxDynamicSharedMemorySize, (int)ldsAtt);
  hipFuncSetAttribute((const void*)attn_kernel<true>,
                      hipFuncAttributeMaxDynamicSharedMemorySize, (int)ldsAtt);

  // 0. Weight prep (transpose + head-major permutation + scale folding)
  prep_qkv_w<<<768, 256, 0, stream>>>(Wqkv_in, bqkv_in, WtQ_in, bpQ_in);
  prep_qkv_w<<<768, 256, 0, stream>>>(Wqkv_out, bqkv_out, WtQ_out, bpQ_out);
  prep_eg_w<<<16, 256, 0, stream>>>(Weg_in, WtE_in);
  prep_eg_w<<<16, 256, 0, stream>>>(Weg_out, WtE_out);
  prep_wo<<<512, 256, 0, stream>>>(Wo, WtO);

  // 1. LayerNorm
  ln_kernel<<<M_TOT / 8, 256, 0, stream>>>(e, ln_g, ln_b, e_ln);

  // 2. Projections (head-major qkv, transposed eg)
  gemm_kernel<false><<<dim3(M_TOT / 16, 6), 256, lds256, stream>>>(e_ln, WtQ_in,  bpQ_in,  qkv_in,  256, 768);
  gemm_kernel<true ><<<dim3(M_TOT / 16, 1), 256, lds256, stream>>>(e_ln, WtE_in,  beg_in,  eg_in_T, 256, 16);
  gemm_kernel<false><<<dim3(M_TOT / 16, 6), 256, lds256, stream>>>(e_ln, WtQ_out, bpQ_out, qkv_out, 256, 768);
  gemm_kernel<true ><<<dim3(M_TOT / 16, 1), 256, lds256, stream>>>(e_ln, WtE_out, beg_out, eg_out_T, 256, 16);

  // 3. Attention: 1280 (j,h) problems per branch, full-S in LDS, TDM staging.
  attn_kernel<false><<<NSEQ * HDIM, 256, ldsAtt, stream>>>(qkv_in,  eg_in_T,  mask, VaBuf);
  attn_kernel<true ><<<NSEQ * HDIM, 256, ldsAtt, stream>>>(qkv_out, eg_out_T, mask, VaBuf);

  // 4. Output projection with matching permuted K-order: [M,512]@[512,256]+bo
  gemm_kernel<false><<<dim3(M_TOT / 16, 2), 256, lds512, stream>>>(VaBuf, WtO, bo, out, 512, 256);
}